// GraphQNNGen318_65481071406927
// MI455X (gfx1250) — compile-verified
//
#include <hip/hip_runtime.h>

typedef float v2f __attribute__((ext_vector_type(2)));
typedef float v8f __attribute__((ext_vector_type(8)));

#define NQ 18
#define NSTATE (1 << NQ)   // 262144 amplitudes per sample
#define BATCH 64
#define NLAYERS 4

// RY matrix [[c,-s],[s,c]] element [mb][kb]
__device__ __forceinline__ float ry_elem(float c, float s, int mb, int kb) {
    return kb ? (mb ? c : -s) : (mb ? s : c);
}

// ---------------------------------------------------------------------------
// Pass 0: closed-form RX encoding of |0..0>  ->  product state.
// amp[b,i] = (-i)^popcount(i) * prod_q (bit_q(i) ? sin(x/2) : cos(x/2))
// Each block covers 2048 consecutive amplitudes (single sample b).
// ---------------------------------------------------------------------------
__global__ __launch_bounds__(256) void init_encode(const float* __restrict__ inputs,
                                                   float2* __restrict__ out) {
    __shared__ float cs[NQ], sn[NQ];
    const unsigned amp0_blk = blockIdx.x * 2048u;
    const int b = amp0_blk >> NQ;
    if (threadIdx.x < NQ) {
        float h = 0.5f * inputs[b * NQ + (int)threadIdx.x];
        cs[threadIdx.x] = cosf(h);
        sn[threadIdx.x] = sinf(h);
    }
    __syncthreads();
    unsigned i0 = (amp0_blk & (NSTATE - 1)) + threadIdx.x * 8u;
    size_t base = ((size_t)b << NQ);
#pragma unroll
    for (int d = 0; d < 8; ++d) {
        unsigned i = i0 + (unsigned)d;
        float r = 1.0f;
#pragma unroll
        for (int q = 0; q < NQ; ++q)
            r *= ((i >> (NQ - 1 - q)) & 1u) ? sn[q] : cs[q];
        int pc = __popc(i) & 3;
        float2 amp;
        amp.x = (pc == 0) ? r : ((pc == 2) ? -r : 0.0f);
        amp.y = (pc == 1) ? -r : ((pc == 3) ? r : 0.0f);
        out[base + i] = amp;
    }
}

// ---------------------------------------------------------------------------
// Fused RY on 4 consecutive qubits [g0..g0+3] via V_WMMA_F32_16X16X4_F32.
// D(16x16) = U(16x16) x S(16x16):  A = U columns chunked K=4, B = state rows.
// One wave per 16x16 tile; 65536 tiles total (1024 per sample).
// State index: i = ((x*16)+k)<<shift | y, shift = 14-g0, column c = x*R+y.
// A layout: lane%16 = M, lane/16 picks K pair {0,1}/{2,3} across 2 VGPRs.
// B layout mirrors A (lane%16 = N). D: vgpr v -> M = v + 8*(lane/16), N = lane%16.
// ---------------------------------------------------------------------------
__global__ __launch_bounds__(256) void ry4_wmma_pass(const float2* __restrict__ in,
                                                     float2* __restrict__ out,
                                                     const float* __restrict__ params,
                                                     int layer, int g0) {
    const int wave = threadIdx.x >> 5;
    const int lane = threadIdx.x & 31;
    const int half = lane >> 4;
    const int n    = lane & 15;
    const int tile = blockIdx.x * 8 + wave;
    const int b    = tile >> 10;
    const int c0   = (tile & 1023) << 4;
    const int shift = 14 - g0;
    const int R     = 1 << shift;

    float c4[4], s4[4];
#pragma unroll
    for (int t = 0; t < 4; ++t) {
        float a = 0.5f * params[layer * NQ + g0 + t];
        c4[t] = cosf(a);
        s4[t] = sinf(a);
    }

    const int c = c0 + n;
    const int x = c >> shift;
    const int y = c & (R - 1);
    const size_t base = ((size_t)b << NQ) + ((size_t)(x << 4) << shift) + (size_t)y;

    v8f accr = {};
    v8f acci = {};
#pragma unroll
    for (int cc = 0; cc < 4; ++cc) {
        const int k = 4 * cc + 2 * half;
        // A chunk: U[n][k], U[n][k+1] as 4-way Kronecker products
        v2f a;
        {
            float p0 = 1.0f, p1 = 1.0f;
#pragma unroll
            for (int t = 0; t < 4; ++t) {
                int mb  = (n >> (3 - t)) & 1;
                int k0b = (k >> (3 - t)) & 1;
                int k1b = ((k + 1) >> (3 - t)) & 1;
                p0 *= ry_elem(c4[t], s4[t], mb, k0b);
                p1 *= ry_elem(c4[t], s4[t], mb, k1b);
            }
            a.x = p0; a.y = p1;
        }
        float2 e0 = in[base + ((size_t)k << shift)];
        float2 e1 = in[base + ((size_t)(k + 1) << shift)];
        v2f br; br.x = e0.x; br.y = e1.x;   // real plane
        v2f bi; bi.x = e0.y; bi.y = e1.y;   // imag plane
        accr = __builtin_amdgcn_wmma_f32_16x16x4_f32(false, a, false, br,
                                                     (short)0, accr, false, false);
        acci = __builtin_amdgcn_wmma_f32_16x16x4_f32(false, a, false, bi,
                                                     (short)0, acci, false, false);
    }
#pragma unroll
    for (int v = 0; v < 8; ++v) {
        int M = v + 8 * half;
        float2 o; o.x = accr[v]; o.y = acci[v];
        out[base + ((size_t)M << shift)] = o;
    }
}

// ---------------------------------------------------------------------------
// RY on qubits 16,17 (4x4 real matrix on 4 contiguous amplitudes) with the
// layer's full CNOT ladder folded in:  out[i] = postRY[i ^ (i>>1)].
// Out-centric: thread owns 4 contiguous outputs; gather block jhi is the
// aligned 4-group of the source, permuted within via low-bit XOR logic.
// ---------------------------------------------------------------------------
__global__ __launch_bounds__(256) void ry2_cnot_pass(const float2* __restrict__ in,
                                                     float2* __restrict__ out,
                                                     const float* __restrict__ params,
                                                     int layer) {
    unsigned tid = blockIdx.x * 256u + threadIdx.x;   // 2^22 threads
    int b = tid >> 16;
    unsigned i0 = (tid & 0xFFFFu) << 2;
    float aA = 0.5f * params[layer * NQ + 16];
    float aB = 0.5f * params[layer * NQ + 17];
    float cA = cosf(aA), sA = sinf(aA);
    float cB = cosf(aB), sB = sinf(aB);

    unsigned jhi = (i0 ^ (i0 >> 1)) & ~3u;
    size_t base = ((size_t)b << NQ);
    float2 vin[4];
#pragma unroll
    for (int k = 0; k < 4; ++k) vin[k] = in[base + jhi + (unsigned)k];

    float2 w[4];
#pragma unroll
    for (int m = 0; m < 4; ++m) {
        float re = 0.0f, im = 0.0f;
#pragma unroll
        for (int k = 0; k < 4; ++k) {
            float u = ry_elem(cA, sA, (m >> 1) & 1, (k >> 1) & 1) *
                      ry_elem(cB, sB, m & 1, k & 1);
            re += u * vin[k].x;
            im += u * vin[k].y;
        }
        w[m].x = re; w[m].y = im;
    }
    int b2 = (i0 >> 2) & 1;
#pragma unroll
    for (int d = 0; d < 4; ++d) {
        int d1 = d >> 1, d0 = d & 1;
        int m = ((d1 ^ b2) << 1) | (d0 ^ d1);   // low 2 bits of i ^ (i>>1)
        out[base + i0 + (unsigned)d] = w[m];
    }
}

// ---------------------------------------------------------------------------
extern "C" void kernel_launch(void* const* d_in, const int* in_sizes, int n_in,
                              void* d_out, int out_size, void* d_ws, size_t ws_size,
                              hipStream_t stream) {
    const float* inputs = (const float*)d_in[0];   // (64, 18) f32
    const float* params = (const float*)d_in[1];   // (4, 18, 1) f32
    float2* S0 = (float2*)d_out;                   // (64, 2^18) complex64
    float2* S1 = (float2*)d_ws;                    // ping buffer (128 MB)

    float2* cur = S0;
    float2* nxt = S1;

    init_encode<<<8192, 256, 0, stream>>>(inputs, cur);
    for (int l = 0; l < NLAYERS; ++l) {
        for (int g0 = 0; g0 <= 12; g0 += 4) {
            ry4_wmma_pass<<<8192, 256, 0, stream>>>(cur, nxt, params, l, g0);
            float2* t = cur; cur = nxt; nxt = t;
        }
        ry2_cnot_pass<<<16384, 256, 0, stream>>>(cur, nxt, params, l);
        float2* t = cur; cur = nxt; nxt = t;
    }
    // 20 gate passes (even) after init -> final state resides in S0 == d_out.
}